// RQS_70806830842451
// MI455X (gfx1250) — compile-verified
//
#include <hip/hip_runtime.h>
#include <hip/hip_bf16.h>
#include <stdint.h>

// ---------------------------------------------------------------------------
// RQS forward (rational-quadratic spline), B*D = 4,194,304 elements, 8 bins.
// Memory-bound: ~435 MB of traffic -> ~19us floor at 23.3 TB/s. Strategy:
//   * w_/h_ : direct global_load_b128 pairs (wave covers contiguous 1KB)
//   * d_    : staged through LDS with the CDNA5 async copy engine
//             (global_load_async_to_lds_b32 + s_wait_asynccnt), then read
//             back conflict-free (stride 7 dwords, 7 odd -> bijective banks)
//   * all transcendentals on hardware paths (v_exp_f32/v_log_f32/v_rcp_f32)
// ---------------------------------------------------------------------------

#define NBINS 8
#define BLOCK 256

__device__ __forceinline__ float softplus_f(float v) {
    // log1p(exp(-|v|)) + max(v,0); log argument is in (1,2] so hardware log
    // is accurate; the 1e-3 derivative floor swamps tail rounding.
    return fmaxf(v, 0.0f) + __logf(1.0f + __expf(-fabsf(v)));
}

__global__ __launch_bounds__(BLOCK) void rqs_fwd_kernel(
    const float* __restrict__ x_,
    const float* __restrict__ w_,
    const float* __restrict__ h_,
    const float* __restrict__ d_,
    float* __restrict__ out,
    int n)
{
    const float TB  = 3.0f;      // tail bound
    const float MBW = 1e-3f;     // min bin width
    const float MBH = 1e-3f;     // min bin height
    const float MD  = 1e-3f;     // min derivative

    __shared__ float s_d[BLOCK * (NBINS - 1)];   // 7 KB per block

    const int t    = threadIdx.x;
    const int base = blockIdx.x * BLOCK;         // first element of this block
    const int i    = base + t;

    // ---- async-stage this block's d_ slice (1792 dwords) into LDS ----------
    // Each wave instruction copies 32 consecutive dwords (128B) straight from
    // global to LDS via the async engine - no VGPR round trip.
    {
        const float*  gsrc = d_ + (size_t)base * (NBINS - 1);
        const uint32_t lds0 = (uint32_t)(uintptr_t)(&s_d[0]);
#pragma unroll
        for (int j = 0; j < NBINS - 1; ++j) {
            const int      idx  = t + BLOCK * j;
            const uint32_t loff = lds0 + (uint32_t)idx * 4u;
            const uint64_t ga   = (uint64_t)(uintptr_t)(gsrc + idx);
            asm volatile("global_load_async_to_lds_b32 %0, %1, off"
                         :: "v"(loff), "v"(ga) : "memory");
        }
        asm volatile("s_wait_asynccnt 0x0\n\t"
                     "s_wait_xcnt 0x0" ::: "memory");
    }
    __syncthreads();

    if (i >= n) return;   // (all blocks full for n = B*D, but keep it safe)

    const size_t i8 = (size_t)i * 8;

    // keep a gfx1250 prefetch in flight for the second half of w_
    __builtin_prefetch(w_ + i8 + 4, 0, 0);

    const float x = x_[i];
    const bool inside = (x >= -TB) && (x <= TB);
    const float xin = fminf(fmaxf(x, -TB), TB);

    // ---- wide coalesced loads: 2x b128 per 8-bin array ----------------------
    const float4 w0 = *reinterpret_cast<const float4*>(w_ + i8);
    const float4 w1 = *reinterpret_cast<const float4*>(w_ + i8 + 4);
    const float4 h0 = *reinterpret_cast<const float4*>(h_ + i8);
    const float4 h1 = *reinterpret_cast<const float4*>(h_ + i8 + 4);

    float wv[NBINS] = {w0.x, w0.y, w0.z, w0.w, w1.x, w1.y, w1.z, w1.w};
    float hv[NBINS] = {h0.x, h0.y, h0.z, h0.w, h1.x, h1.y, h1.z, h1.w};

    // d_ values from LDS: address t*28 + 4k -> bank (7t+k) mod 64, conflict-free
    float dv[NBINS - 1];
#pragma unroll
    for (int k = 0; k < NBINS - 1; ++k) dv[k] = s_d[t * (NBINS - 1) + k];

    // ---- widths: softmax -> affine -> cumsum -> forced endpoints ------------
    float cw[NBINS + 1];
    {
        float m = wv[0];
#pragma unroll
        for (int k = 1; k < NBINS; ++k) m = fmaxf(m, wv[k]);
        float e[NBINS];
        float s = 0.0f;
#pragma unroll
        for (int k = 0; k < NBINS; ++k) { e[k] = __expf(wv[k] - m); s += e[k]; }
        const float inv = __builtin_amdgcn_rcpf(s);
        const float scale = (1.0f - MBW * (float)NBINS);
        float cs = 0.0f;
        cw[0] = -TB;
#pragma unroll
        for (int k = 0; k < NBINS; ++k) {
            const float wid = MBW + scale * (e[k] * inv);
            cs += wid;
            cw[k + 1] = 6.0f * cs - 3.0f;    // (right-left)*cumsum + left
        }
        cw[NBINS] = TB;                       // exact right endpoint
    }

    // ---- heights: same construction -----------------------------------------
    float ch[NBINS + 1];
    {
        float m = hv[0];
#pragma unroll
        for (int k = 1; k < NBINS; ++k) m = fmaxf(m, hv[k]);
        float e[NBINS];
        float s = 0.0f;
#pragma unroll
        for (int k = 0; k < NBINS; ++k) { e[k] = __expf(hv[k] - m); s += e[k]; }
        const float inv = __builtin_amdgcn_rcpf(s);
        const float scale = (1.0f - MBH * (float)NBINS);
        float cs = 0.0f;
        ch[0] = -TB;
#pragma unroll
        for (int k = 0; k < NBINS; ++k) {
            const float hei = MBH + scale * (e[k] * inv);
            cs += hei;
            ch[k + 1] = 6.0f * cs - 3.0f;
        }
        ch[NBINS] = TB;
    }

    // ---- derivatives: pad so boundary derivative == 1 exactly ---------------
    float dd[NBINS + 1];
    dd[0]     = 1.0f;   // MD + softplus(log(exp(1-MD)-1)) == 1.0 in f32
    dd[NBINS] = 1.0f;
#pragma unroll
    for (int k = 0; k < NBINS - 1; ++k) dd[k + 1] = MD + softplus_f(dv[k]);

    // ---- branchless bin search fused with the gathers -----------------------
    // bin = max k in [0,7] with xin >= cw[k]  (== reference count-1, clipped)
    float in_cw = cw[0], cwn = cw[1];
    float in_ch = ch[0], chn = ch[1];
    float dk = dd[0], dk1 = dd[1];
#pragma unroll
    for (int k = 1; k < NBINS; ++k) {
        const bool g = (xin >= cw[k]);
        in_cw = g ? cw[k]     : in_cw;
        cwn   = g ? cw[k + 1] : cwn;
        in_ch = g ? ch[k]     : in_ch;
        chn   = g ? ch[k + 1] : chn;
        dk    = g ? dd[k]     : dk;
        dk1   = g ? dd[k + 1] : dk1;
    }

    // ---- rational-quadratic spline ------------------------------------------
    const float in_w  = cwn - in_cw;     // widths recomputed from cum diffs
    const float in_h  = chn - in_ch;
    const float invw  = __builtin_amdgcn_rcpf(in_w);   // in_w >= ~6e-3 > 0
    const float delta = in_h * invw;
    const float theta = (xin - in_cw) * invw;
    const float omt   = 1.0f - theta;
    const float t1m   = theta * omt;
    const float th2   = theta * theta;

    const float numer  = in_h * (delta * th2 + dk * t1m);
    const float denom  = delta + (dk + dk1 - 2.0f * delta) * t1m;
    const float invden = __builtin_amdgcn_rcpf(denom);
    const float y      = in_ch + numer * invden;

    const float dnum = (delta * delta) *
                       (dk1 * th2 + 2.0f * delta * t1m + dk * omt * omt);
    const float lad  = __logf(dnum) - 2.0f * __logf(denom);

    out[i]     = inside ? y   : x;       // identity tails
    out[n + i] = inside ? lad : 0.0f;
}

extern "C" void kernel_launch(void* const* d_in, const int* in_sizes, int n_in,
                              void* d_out, int out_size, void* d_ws, size_t ws_size,
                              hipStream_t stream) {
    const float* x_ = (const float*)d_in[0];   // (B, D)
    const float* w_ = (const float*)d_in[1];   // (B, D, 8)
    const float* h_ = (const float*)d_in[2];   // (B, D, 8)
    const float* d_ = (const float*)d_in[3];   // (B, D, 7)
    float* out = (float*)d_out;                // [outputs | logabsdet]

    const int n = in_sizes[0];                 // B*D = 4,194,304
    const int blocks = (n + BLOCK - 1) / BLOCK;
    rqs_fwd_kernel<<<blocks, BLOCK, 0, stream>>>(x_, w_, h_, d_, out, n);
}